// GraphSAGENetwork_55946243997754
// MI455X (gfx1250) — compile-verified
//
#include <hip/hip_runtime.h>
#include <math.h>

typedef __attribute__((ext_vector_type(2))) float v2f;
typedef __attribute__((ext_vector_type(8))) float v8f;

#define N_NODES 100000
#define SAGE_EPS 1e-12f

// ---------------------------------------------------------------------------
// deg[d] += 1 per edge (in-degree)
// ---------------------------------------------------------------------------
__global__ void degree_kernel(const int* __restrict__ dst,
                              float* __restrict__ deg, int nEdges) {
  int e = blockIdx.x * blockDim.x + threadIdx.x;
  if (e < nEdges) atomicAdd(&deg[dst[e]], 1.0f);
}

// ---------------------------------------------------------------------------
// accum[dst] += feat[src]  (scatter-add, CHUNKS threads per edge)
// feat table is L2-resident (<=12.8MB); atomics resolve at L2.
// ---------------------------------------------------------------------------
template <int F, int CHUNKS>
__global__ void scatter_add_kernel(const float* __restrict__ feat,
                                   const int* __restrict__ src,
                                   const int* __restrict__ dst,
                                   float* __restrict__ accum, int nEdges) {
  long long tid = (long long)blockIdx.x * blockDim.x + threadIdx.x;
  long long total = (long long)nEdges * CHUNKS;
  if (tid >= total) return;
  int e = (int)(tid / CHUNKS);
  int c = (int)(tid % CHUNKS);
  int s = src[e];
  int d = dst[e];
  constexpr int FP = F / CHUNKS;
  const float* __restrict__ frow = feat + (long long)s * F + c * FP;
  float* __restrict__ orow = accum + (long long)d * F + c * FP;
#pragma unroll
  for (int f = 0; f < FP; ++f) atomicAdd(&orow[f], frow[f]);
}

// ---------------------------------------------------------------------------
// Fused SAGEConv dense stage, one wave32 per 16 nodes:
//   out = relu(l2norm_row( (agg/max(deg,1)) @ Wl + feat @ Wr + bl ))
// Two separate K-chains (agg@Wl, feat@Wr), each K padded to a multiple of 4,
// stepped with V_WMMA_F32_16X16X4_F32. Padding is branchless: clamp the index
// to an always-valid element and select the per-element scale (dinv/1/0).
// A layout  (16x4 f32): lane l holds row (l&15), K = 2*(l>>4)+{0,1}
// B layout  (4x16 f32): lane l holds col (l&15), K = 2*(l>>4)+{0,1}
// C/D layout(16x16 f32): vgpr i -> row i+8*(l>>4), col (l&15)
// Weights are staged in LDS once per block (B-fragments become ds_loads).
// ---------------------------------------------------------------------------
template <int KIN>
__global__ void sage_dense_wmma(const float* __restrict__ agg,
                                const float* __restrict__ feat,
                                const float* __restrict__ deg,
                                const float* __restrict__ Wl,
                                const float* __restrict__ bl,
                                const float* __restrict__ Wr,
                                float* __restrict__ out, int nNodes) {
  constexpr int KPAD = (KIN + 3) & ~3;
  constexpr int KSTEPS = KPAD / 4;

  __shared__ float sWl[KIN * 32];
  __shared__ float sWr[KIN * 32];
  for (int i = threadIdx.x; i < KIN * 32; i += blockDim.x) {
    sWl[i] = Wl[i];
    sWr[i] = Wr[i];
  }
  __syncthreads();  // all waves reach this before any early exit

  const int lane = threadIdx.x & 31;
  const int wave = (int)((blockIdx.x * (long long)blockDim.x + threadIdx.x) >> 5);
  const int m0 = wave * 16;
  if (m0 >= nNodes) return;

  const int half = lane >> 4;  // which K-pair / row-half this lane serves
  const int l16 = lane & 15;

  int rowA = m0 + l16;
  if (rowA >= nNodes) rowA = nNodes - 1;  // clamp loads; stores are guarded
  const float dinv = 1.0f / fmaxf(deg[rowA], 1.0f);
  const float* __restrict__ arow = agg + (long long)rowA * KIN;
  const float* __restrict__ frow = feat + (long long)rowA * KIN;

  const int n0 = l16;       // column for accumulator c0
  const int n1 = 16 + l16;  // column for accumulator c1

  v8f c0 = {};
  v8f c1 = {};

#pragma unroll
  for (int s = 0; s < KSTEPS; ++s) {
    const int ka = 4 * s + 2 * half;
    // Branchless pad handling (compile-time no-op when KIN % 4 == 0)
    const bool u0 = (ka + 0) < KIN;
    const bool u1 = (ka + 1) < KIN;
    const int k0 = u0 ? (ka + 0) : 0;
    const int k1 = u1 ? (ka + 1) : 0;

    // ---- chain 1: (agg * dinv) @ Wl ----
    v2f a, b0, b1;
    a.x = arow[k0] * (u0 ? dinv : 0.0f);
    a.y = arow[k1] * (u1 ? dinv : 0.0f);
    b0.x = sWl[k0 * 32 + n0];
    b0.y = sWl[k1 * 32 + n0];
    b1.x = sWl[k0 * 32 + n1];
    b1.y = sWl[k1 * 32 + n1];
    c0 = __builtin_amdgcn_wmma_f32_16x16x4_f32(false, a, false, b0, (short)0, c0,
                                               false, false);
    c1 = __builtin_amdgcn_wmma_f32_16x16x4_f32(false, a, false, b1, (short)0, c1,
                                               false, false);

    // ---- chain 2: feat @ Wr ----
    a.x = frow[k0] * (u0 ? 1.0f : 0.0f);
    a.y = frow[k1] * (u1 ? 1.0f : 0.0f);
    b0.x = sWr[k0 * 32 + n0];
    b0.y = sWr[k1 * 32 + n0];
    b1.x = sWr[k0 * 32 + n1];
    b1.y = sWr[k1 * 32 + n1];
    c0 = __builtin_amdgcn_wmma_f32_16x16x4_f32(false, a, false, b0, (short)0, c0,
                                               false, false);
    c1 = __builtin_amdgcn_wmma_f32_16x16x4_f32(false, a, false, b1, (short)0, c1,
                                               false, false);
  }

  const float bias0 = bl[n0];
  const float bias1 = bl[n1];

#pragma unroll
  for (int i = 0; i < 8; ++i) {
    const int r = m0 + i + 8 * half;
    float v0 = c0[i] + bias0;
    float v1 = c1[i] + bias1;
    // L2 row norm: reduce v0^2+v1^2 across the 16-lane half (bit4 never flips)
    float ss = v0 * v0 + v1 * v1;
    ss += __shfl_xor(ss, 1, 32);
    ss += __shfl_xor(ss, 2, 32);
    ss += __shfl_xor(ss, 4, 32);
    ss += __shfl_xor(ss, 8, 32);
    const float scale = 1.0f / fmaxf(sqrtf(ss), SAGE_EPS);
    v0 = fmaxf(v0 * scale, 0.0f);  // relu
    v1 = fmaxf(v1 * scale, 0.0f);
    if (r < nNodes) {
      out[(long long)r * 32 + n0] = v0;
      out[(long long)r * 32 + n1] = v1;
    }
  }
}

// ---------------------------------------------------------------------------
// Final head: out[N,2] = h[N,32] @ Wlin[32,2] + blin   (tiny; VALU is fine)
// ---------------------------------------------------------------------------
__global__ void final_linear_kernel(const float* __restrict__ h,
                                    const float* __restrict__ Wlin,
                                    const float* __restrict__ blin,
                                    float* __restrict__ out, int nNodes) {
  int i = blockIdx.x * blockDim.x + threadIdx.x;
  if (i >= nNodes) return;
  const float* __restrict__ hr = h + (long long)i * 32;
  float a0 = blin[0], a1 = blin[1];
#pragma unroll
  for (int k = 0; k < 32; ++k) {
    float hv = hr[k];
    a0 += hv * Wlin[k * 2 + 0];
    a1 += hv * Wlin[k * 2 + 1];
  }
  out[(long long)i * 2 + 0] = a0;
  out[(long long)i * 2 + 1] = a1;
}

// ---------------------------------------------------------------------------
extern "C" void kernel_launch(void* const* d_in, const int* in_sizes, int n_in,
                              void* d_out, int out_size, void* d_ws, size_t ws_size,
                              hipStream_t stream) {
  const float* x    = (const float*)d_in[0];
  const int*   ei   = (const int*)d_in[1];
  const float* W1l  = (const float*)d_in[2];
  const float* b1   = (const float*)d_in[3];
  const float* W1r  = (const float*)d_in[4];
  const float* W2l  = (const float*)d_in[5];
  const float* b2   = (const float*)d_in[6];
  const float* W2r  = (const float*)d_in[7];
  const float* Wlin = (const float*)d_in[8];
  const float* blin = (const float*)d_in[9];

  const int nEdges = in_sizes[1] / 2;
  const int* src = ei;           // edge_index[0]
  const int* dst = ei + nEdges;  // edge_index[1]
  const int N = N_NODES;

  // Workspace partition (256B aligned slabs)
  char* ws = (char*)d_ws;
  size_t off = 0;
  auto alloc = [&](size_t bytes) -> char* {
    char* p = ws + off;
    off = (off + bytes + 255) & ~(size_t)255;
    return p;
  };
  float* deg  = (float*)alloc((size_t)N * sizeof(float));
  float* agg1 = (float*)alloc((size_t)N * 14 * sizeof(float));
  float* h1   = (float*)alloc((size_t)N * 32 * sizeof(float));
  float* agg2 = (float*)alloc((size_t)N * 32 * sizeof(float));
  float* h2   = (float*)alloc((size_t)N * 32 * sizeof(float));
  (void)ws_size;

  // Zero accumulators every call (graph-capturable memset nodes)
  hipMemsetAsync(deg, 0, (size_t)N * sizeof(float), stream);
  hipMemsetAsync(agg1, 0, (size_t)N * 14 * sizeof(float), stream);
  hipMemsetAsync(agg2, 0, (size_t)N * 32 * sizeof(float), stream);

  const int T = 256;

  // In-degree (shared by both layers)
  degree_kernel<<<(nEdges + T - 1) / T, T, 0, stream>>>(dst, deg, nEdges);

  // Layer 1 aggregation: agg1[dst] += x[src], 2 threads/edge
  {
    long long total = (long long)nEdges * 2;
    scatter_add_kernel<14, 2><<<(int)((total + T - 1) / T), T, 0, stream>>>(
        x, src, dst, agg1, nEdges);
  }
  // Layer 1 dense (WMMA fp32) -> h1
  {
    int waves = (N + 15) / 16;
    int wpb = T / 32;
    sage_dense_wmma<14><<<(waves + wpb - 1) / wpb, T, 0, stream>>>(
        agg1, x, deg, W1l, b1, W1r, h1, N);
  }
  // Layer 2 aggregation: agg2[dst] += h1[src], 4 threads/edge
  {
    long long total = (long long)nEdges * 4;
    scatter_add_kernel<32, 4><<<(int)((total + T - 1) / T), T, 0, stream>>>(
        h1, src, dst, agg2, nEdges);
  }
  // Layer 2 dense (WMMA fp32) -> h2
  {
    int waves = (N + 15) / 16;
    int wpb = T / 32;
    sage_dense_wmma<32><<<(waves + wpb - 1) / wpb, T, 0, stream>>>(
        agg2, h1, deg, W2l, b2, W2r, h2, N);
  }
  // Head
  final_linear_kernel<<<(N + T - 1) / T, T, 0, stream>>>(h2, Wlin, blin,
                                                         (float*)d_out, N);
}